// spatial_attention_56822417326178
// MI455X (gfx1250) — compile-verified
//
#include <hip/hip_runtime.h>

typedef float f4  __attribute__((ext_vector_type(4)));
typedef int   v4i __attribute__((ext_vector_type(4)));

#define B_  16
#define C_  64
#define H_  256
#define W_  256
#define HW_ (H_ * W_)
#define ROWS_ 4   // rows of output per block in pass 2

// gfx1250 async global->LDS DMA path (ASYNCcnt); plain LDS store fallback.
#if defined(__gfx1250__) && __has_builtin(__builtin_amdgcn_global_load_async_to_lds_b128)
#define USE_ASYNC_LDS 1
typedef __attribute__((address_space(1))) v4i* gptr_v4i;  // global
typedef __attribute__((address_space(3))) v4i* lptr_v4i;  // LDS
#else
#define USE_ASYNC_LDS 0
#endif

// ---------------------------------------------------------------------------
// Pass 1: per-pixel channel mean & max over C=64.
// Each thread owns 4 consecutive W positions (one b128 quad) and loops the 64
// channels with non-temporal b128 loads (x is pure streaming traffic).
// s layout: [B][2][H][W]  (channel 0 = avg, channel 1 = max)
// ---------------------------------------------------------------------------
__global__ __launch_bounds__(256) void sa_reduce_kernel(
    const float* __restrict__ x, float* __restrict__ s) {
  const int q   = blockIdx.x * 256 + threadIdx.x;   // quad index in [0, B*HW/4)
  const int qpb = HW_ / 4;                          // quads per image plane
  const int b   = q / qpb;
  const int p4  = (q - b * qpb) * 4;                // pixel offset within plane

  const float* base = x + (size_t)b * C_ * HW_ + p4;

  f4 sum = {0.f, 0.f, 0.f, 0.f};
  const float ninf = -__builtin_inff();
  f4 mx = {ninf, ninf, ninf, ninf};

#pragma unroll 8
  for (int c = 0; c < C_; ++c) {
    f4 v = __builtin_nontemporal_load((const f4*)(base + (size_t)c * HW_));
    sum += v;
    mx.x = fmaxf(mx.x, v.x);
    mx.y = fmaxf(mx.y, v.y);
    mx.z = fmaxf(mx.z, v.z);
    mx.w = fmaxf(mx.w, v.w);
  }

  f4 avg = sum * (1.0f / (float)C_);
  float* sp = s + (size_t)b * 2 * HW_ + p4;
  *(f4*)sp         = avg;   // regular (temporal) store: keep s resident in L2
  *(f4*)(sp + HW_) = mx;
}

// ---------------------------------------------------------------------------
// Pass 2: 3x3 conv over s (+pad 1) -> sigmoid -> multiply into x.
// Block = 256 threads handles (b, rows h0..h0+3, all 256 W).
// Stages s rows h0-1..h0+4 (2 ch x 6 rows x 256 f32 = 12KB) into LDS via the
// async DMA path, then each thread computes 4 attention values and streams
// 64 channels of x through a NT load / mul / NT store loop.
// ---------------------------------------------------------------------------
__global__ __launch_bounds__(256) void sa_conv_apply_kernel(
    const float* __restrict__ x, const float* __restrict__ s,
    const float* __restrict__ cw, float* __restrict__ out) {
  __shared__ float sb[2][6][W_];   // [channel][row h0-1+rr][w]  (12 KB)

  const int tid = threadIdx.x;
  const int hpb = H_ / ROWS_;                 // 64 row-groups per image
  const int b   = blockIdx.x / hpb;
  const int h0  = (blockIdx.x - b * hpb) * ROWS_;

  // ---- stage s tile into LDS: 2ch * 6 rows * 64 quads = 768 b128 loads ----
#pragma unroll
  for (int i = 0; i < 3; ++i) {
    const int li     = tid + i * 256;         // [0,768)
    const int rowIdx = li >> 6;               // [0,12)
    const int qq     = li & 63;
    const int ch     = (rowIdx >= 6) ? 1 : 0;
    const int rr     = rowIdx - ch * 6;
    const int hh     = h0 - 1 + rr;
    if (hh >= 0 && hh < H_) {
      const float* gp = s + (size_t)(b * 2 + ch) * HW_ + (size_t)hh * W_ + qq * 4;
      float*       lp = &sb[ch][rr][qq * 4];
#if USE_ASYNC_LDS
      __builtin_amdgcn_global_load_async_to_lds_b128(
          (gptr_v4i)gp, (lptr_v4i)lp, /*offset=*/0, /*cpol=*/0);
#else
      *(f4*)lp = *(const f4*)gp;
#endif
    }
    // invalid (out-of-image) rows are never read: the conv guards on hh.
  }
#if USE_ASYNC_LDS
#if __has_builtin(__builtin_amdgcn_s_wait_asynccnt)
  __builtin_amdgcn_s_wait_asynccnt(0);
#else
  asm volatile("s_wait_asynccnt 0" ::: "memory");
#endif
#endif
  __syncthreads();

  // ---- conv weights (uniform; 18 scalars) ----
  float wv[18];
#pragma unroll
  for (int i = 0; i < 18; ++i) wv[i] = cw[i];

  // ---- 3x3 conv + sigmoid for this thread's 4 output pixels ----
  const int r  = tid >> 6;        // row within block: 0..3
  const int qw = tid & 63;        // quad within row
  const int h  = h0 + r;
  const int w0 = qw * 4;

  f4 att;
#pragma unroll
  for (int j = 0; j < 4; ++j) {
    const int w = w0 + j;
    float acc = 0.f;
#pragma unroll
    for (int ky = 0; ky < 3; ++ky) {
      const int hh = h + ky - 1;
      if (hh < 0 || hh >= H_) continue;   // zero padding (rows)
      const int rr = r + ky;              // LDS row = (hh - (h0-1))
#pragma unroll
      for (int kx = 0; kx < 3; ++kx) {
        const int ww = w + kx - 1;
        if (ww < 0 || ww >= W_) continue; // zero padding (cols)
        acc = fmaf(wv[ky * 3 + kx],     sb[0][rr][ww], acc);
        acc = fmaf(wv[9 + ky * 3 + kx], sb[1][rr][ww], acc);
      }
    }
    att[j] = 1.0f / (1.0f + expf(-acc));
  }

  // ---- stream 64 channels of x through NT load / mul / NT store ----
  const size_t base = (size_t)b * C_ * HW_ + (size_t)h * W_ + w0;
  const float* xb = x + base;
  float*       ob = out + base;
#pragma unroll 4
  for (int c = 0; c < C_; ++c) {
    f4 xv = __builtin_nontemporal_load((const f4*)(xb + (size_t)c * HW_));
    f4 ov = xv * att;
    __builtin_nontemporal_store(ov, (f4*)(ob + (size_t)c * HW_));
  }
}

// ---------------------------------------------------------------------------
extern "C" void kernel_launch(void* const* d_in, const int* in_sizes, int n_in,
                              void* d_out, int out_size, void* d_ws, size_t ws_size,
                              hipStream_t stream) {
  (void)in_sizes; (void)n_in; (void)out_size; (void)ws_size;
  const float* x  = (const float*)d_in[0];   // [16,64,256,256]
  const float* cw = (const float*)d_in[1];   // [1,2,3,3]
  float*       o  = (float*)d_out;           // [16,64,256,256]
  float*       s  = (float*)d_ws;            // [16,2,256,256] = 8 MB scratch

  const int quads = B_ * HW_ / 4;                        // 262,144
  sa_reduce_kernel<<<quads / 256, 256, 0, stream>>>(x, s);

  const int blocks2 = B_ * (H_ / ROWS_);                 // 1,024
  sa_conv_apply_kernel<<<blocks2, 256, 0, stream>>>(x, s, cw, o);
}